// AnchorTarget_27015344291857
// MI455X (gfx1250) — compile-verified
//
#include <hip/hip_runtime.h>
#include <hip/hip_bf16.h>
#include <stdint.h>

// ---------------- problem constants (from reference setup) ----------------
#define RR      128
#define CC      128
#define NA      9
#define NANCH   (RR*CC*NA)      // 147456 anchors
#define NGT     128             // gt boxes
#define NEG_OV  0.3f
#define POS_OV  0.7f
#define RPN_BATCH 256
#define MAX_POS   128

typedef unsigned int uint32x4 __attribute__((ext_vector_type(4)));
typedef int          int32x8  __attribute__((ext_vector_type(8)));
typedef int          int32x4  __attribute__((ext_vector_type(4)));

// Base anchor (W,H): ratios (0.5,1,2) x scales (8,16,32), Python round() semantics.
__device__ __constant__ float kW[NA] = {184.f,368.f,736.f,128.f,256.f,512.f, 88.f,176.f,352.f};
__device__ __constant__ float kH[NA] = { 96.f,192.f,384.f,128.f,256.f,512.f,176.f,352.f,704.f};

// ---------------- CDNA5 TDM staging: global -> LDS via tensor_load_to_lds ----
// Builds a D# (groups 0..3) for a 1-row tile of `nfloats` 4-byte elements and
// issues the Tensor Data Mover DMA from wave 0; waits on TENSORcnt, then
// workgroup barrier so all waves see the staged table.
__device__ __forceinline__ void tdm_stage_gt(const float* __restrict__ src,
                                             float* sdst, int nfloats) {
    if (threadIdx.x == 0) {
        unsigned long long ga = (unsigned long long)(uintptr_t)src;
        unsigned lds = (unsigned)(uintptr_t)sdst;   // low 32 bits of flat = LDS byte offset
        uint32x4 g0;
        g0.x = 1u;                                   // count = 1 (valid descriptor)
        g0.y = lds;                                  // lds_addr
        g0.z = (unsigned)ga;                         // global_addr[31:0]
        g0.w = ((unsigned)(ga >> 32) & 0x01FFFFFFu)  // global_addr[56:32]
             | (2u << 30);                           // type = 2 ("image")
        int32x8 g1;
        g1[0] = 0x00020000;                          // workgroup_mask=0, data_size=4B
        g1[1] = (int)((unsigned)nfloats << 16);      // tensor_dim0[15:0]
        g1[2] = (1 << 16) | ((nfloats >> 16) & 0xFFFF); // tensor_dim0 hi | tensor_dim1=1
        g1[3] = (int)((unsigned)nfloats << 16);      // tile_dim0
        g1[4] = 1;                                   // tile_dim1=1, tile_dim2=0
        g1[5] = nfloats;                             // tensor_dim0_stride lo
        g1[6] = 0;
        g1[7] = 0;
        int32x4 gz = {0, 0, 0, 0};                   // groups 2/3 unused (zeros)
        asm volatile("tensor_load_to_lds %0, %1, %2, %3"
                     :: "s"(g0), "s"(g1), "s"(gz), "s"(gz)
                     : "memory");
        __builtin_amdgcn_s_wait_tensorcnt(0);        // TENSORcnt == 0 -> data in LDS
    }
    __syncthreads();
}

// anchor n -> corners
__device__ __forceinline__ void anchor_coords(int n, float& ax1, float& ay1,
                                              float& ax2, float& ay2,
                                              float& W, float& H) {
    int a = n % NA;
    int k = n / NA;
    int j = k % RR;           // shift_x index
    int i = k / RR;           // shift_y index
    float sx = (float)j * 16.0f;
    float sy = (float)i * 16.0f;
    W = kW[a]; H = kH[a];
    ax1 = sx + 7.5f - 0.5f * (W - 1.0f);
    ay1 = sy + 7.5f - 0.5f * (H - 1.0f);
    ax2 = sx + 7.5f + 0.5f * (W - 1.0f);
    ay2 = sy + 7.5f + 0.5f * (H - 1.0f);
}

// ---------------- kernel 0: init per-gt column-max slots ----------------
// key = (float_bits(iou) << 32) | (~anchor_idx)  -> unsigned max == argmax with
// lowest-index tie-break. Init = iou 0.0 @ idx 0 (matches argmax of tied column).
__global__ void k_init(unsigned long long* __restrict__ slots) {
    if (threadIdx.x < NGT) slots[threadIdx.x] = 0xFFFFFFFFull;
}

// ---------------- kernel 1: IoU row-max/argmax + column atomic argmax ----
__global__ void k_overlap(const float* __restrict__ gt,
                          const float* __restrict__ meta,
                          float* __restrict__ maxov,
                          int* __restrict__ ainfo,
                          unsigned long long* __restrict__ slots) {
    __shared__ float sGT[NGT * 4];
    __shared__ float sAreaB[NGT];
    tdm_stage_gt(gt, sGT, NGT * 4);
    if (threadIdx.x < NGT) {
        int g = threadIdx.x;
        float bx1 = sGT[g*4+0], by1 = sGT[g*4+1], bx2 = sGT[g*4+2], by2 = sGT[g*4+3];
        sAreaB[g] = (bx2 - bx1 + 1.0f) * (by2 - by1 + 1.0f);
    }
    __syncthreads();

    int n = blockIdx.x * blockDim.x + threadIdx.x;
    if (n >= NANCH) return;

    float ax1, ay1, ax2, ay2, W, H;
    anchor_coords(n, ax1, ay1, ax2, ay2, W, H);
    float height = meta[0], width = meta[1];
    bool inside = (ax1 >= 0.0f) && (ay1 >= 0.0f) && (ax2 < width) && (ay2 < height);

    float best = -1.0f;     // outside anchors: masked overlaps = -1 everywhere
    int   barg = 0;
    if (inside) {
        float areaA = W * H;                    // (x2-x1+1)*(y2-y1+1) exactly
        unsigned invn = 0xFFFFFFFFu - (unsigned)n;
        #pragma unroll 4
        for (int g = 0; g < NGT; ++g) {
            float bx1 = sGT[g*4+0], by1 = sGT[g*4+1];
            float bx2 = sGT[g*4+2], by2 = sGT[g*4+3];
            float iw = fminf(ax2, bx2) - fmaxf(ax1, bx1) + 1.0f;
            float ih = fminf(ay2, by2) - fmaxf(ay1, by1) + 1.0f;
            iw = fmaxf(iw, 0.0f);
            ih = fmaxf(ih, 0.0f);
            float inter = iw * ih;
            float iou = inter / (areaA + sAreaB[g] - inter);
            if (iou > best) { best = iou; barg = g; }   // strict > == first-max tie rule
            if (iou > 0.0f) {                           // prune: zero-IoU never beats init
                unsigned long long key =
                    ((unsigned long long)__float_as_uint(iou) << 32) | invn;
                atomicMax(&slots[g], key);
            }
        }
    }
    maxov[n] = best;
    ainfo[n] = barg | (inside ? (1 << 8) : 0);
}

// ---------------- kernel 2: labels + regression targets ----------------
__global__ void k_labels_targets(const float* __restrict__ gt,
                                 const float* __restrict__ maxov,
                                 const int* __restrict__ ainfo,
                                 float* __restrict__ out) {
    __shared__ float sGT[NGT * 4];
    tdm_stage_gt(gt, sGT, NGT * 4);

    int n = blockIdx.x * blockDim.x + threadIdx.x;
    if (n >= NANCH) return;

    float mo = maxov[n];
    int info = ainfo[n];
    bool inside = (info >> 8) & 1;
    int g = info & 0xFF;

    float lab = -1.0f;
    if (inside && mo < NEG_OV) lab = 0.0f;
    if (mo >= POS_OV)          lab = 1.0f;
    out[n] = lab;

    float* tgt = out + NANCH;
    if (!inside) {
        tgt[n*4+0] = 0.0f; tgt[n*4+1] = 0.0f; tgt[n*4+2] = 0.0f; tgt[n*4+3] = 0.0f;
    } else {
        float ax1, ay1, ax2, ay2, W, H;
        anchor_coords(n, ax1, ay1, ax2, ay2, W, H);
        float gx1 = sGT[g*4+0], gy1 = sGT[g*4+1], gx2 = sGT[g*4+2], gy2 = sGT[g*4+3];
        float gw  = gx2 - gx1 + 1.0f;
        float gh  = gy2 - gy1 + 1.0f;
        float gcx = gx1 + 0.5f * gw;
        float gcy = gy1 + 0.5f * gh;
        float ecx = ax1 + 0.5f * W;
        float ecy = ay1 + 0.5f * H;
        tgt[n*4+0] = (gcx - ecx) / W;
        tgt[n*4+1] = (gcy - ecy) / H;
        tgt[n*4+2] = logf(gw / W);
        tgt[n*4+3] = logf(gh / H);
    }
}

// ---------------- kernel 3: labels[gt_argmax] = 1 ----------------
__global__ void k_scatter(const unsigned long long* __restrict__ slots,
                          float* __restrict__ labels) {
    int g = threadIdx.x;
    if (g < NGT) {
        unsigned inv = (unsigned)(slots[g] & 0xFFFFFFFFull);
        unsigned idx = 0xFFFFFFFFu - inv;
        if (idx < (unsigned)NANCH) labels[idx] = 1.0f;
    }
}

// ---------------- kernel 4: capped subsampling (single block scan) ------
__global__ void k_subsample(float* __restrict__ labels) {
    __shared__ int s[1024];
    __shared__ int running;
    __shared__ int cap2;
    int tid = threadIdx.x;

    // pass 1: cap positives at MAX_POS (index-rank deterministic selection)
    if (tid == 0) running = 0;
    __syncthreads();
    for (int base = 0; base < NANCH; base += 1024) {
        int n = base + tid;
        int flag = (n < NANCH && labels[n] == 1.0f) ? 1 : 0;
        s[tid] = flag;
        __syncthreads();
        for (int off = 1; off < 1024; off <<= 1) {
            int v = (tid >= off) ? s[tid - off] : 0;
            __syncthreads();
            s[tid] += v;
            __syncthreads();
        }
        int rank = running + s[tid] - flag;   // exclusive rank among positives
        if (flag && rank >= MAX_POS && n < NANCH) labels[n] = -1.0f;
        __syncthreads();
        if (tid == 0) running += s[1023];
        __syncthreads();
    }
    if (tid == 0) {
        int kept = running < MAX_POS ? running : MAX_POS;
        cap2 = RPN_BATCH - kept;
        running = 0;
    }
    __syncthreads();

    // pass 2: cap negatives at 256 - n_pos
    for (int base = 0; base < NANCH; base += 1024) {
        int n = base + tid;
        int flag = (n < NANCH && labels[n] == 0.0f) ? 1 : 0;
        s[tid] = flag;
        __syncthreads();
        for (int off = 1; off < 1024; off <<= 1) {
            int v = (tid >= off) ? s[tid - off] : 0;
            __syncthreads();
            s[tid] += v;
            __syncthreads();
        }
        int rank = running + s[tid] - flag;
        if (flag && rank >= cap2 && n < NANCH) labels[n] = -1.0f;
        __syncthreads();
        if (tid == 0) running += s[1023];
        __syncthreads();
    }
}

// ---------------- launcher ----------------
extern "C" void kernel_launch(void* const* d_in, const int* in_sizes, int n_in,
                              void* d_out, int out_size, void* d_ws, size_t ws_size,
                              hipStream_t stream) {
    (void)in_sizes; (void)n_in; (void)out_size; (void)ws_size;
    // d_in[0] = scores (shape-only), d_in[1] = gt_boxes (128x4), d_in[2] = metadata (3)
    const float* gt   = (const float*)d_in[1];
    const float* meta = (const float*)d_in[2];
    float* out = (float*)d_out;

    char* ws = (char*)d_ws;
    unsigned long long* slots = (unsigned long long*)ws;              // 1 KB
    float* maxov = (float*)(ws + 1024);                               // N*4 B
    int*   ainfo = (int*)(ws + 1024 + (size_t)NANCH * 4);             // N*4 B

    const int threads = 256;
    const int blocks  = (NANCH + threads - 1) / threads;              // 576

    k_init<<<1, NGT, 0, stream>>>(slots);
    k_overlap<<<blocks, threads, 0, stream>>>(gt, meta, maxov, ainfo, slots);
    k_labels_targets<<<blocks, threads, 0, stream>>>(gt, maxov, ainfo, out);
    k_scatter<<<1, NGT, 0, stream>>>(slots, out);
    k_subsample<<<1, 1024, 0, stream>>>(out);
}